// FIoUCriterion_78881369358863
// MI455X (gfx1250) — compile-verified
//
#include <hip/hip_runtime.h>
#include <hip/hip_bf16.h>

typedef __attribute__((ext_vector_type(2))) float v2f;
typedef __attribute__((ext_vector_type(8))) float v8f;

#define NB 32          // batches
#define NN 64          // nodes
#define HW 16384       // h*w
#define KC 16          // K-chunks per batch (grid split)
#define KCHUNK (HW / KC)   // 1024
#define KS 64          // K staged in LDS per iteration
#define LDS_STRIDE 68  // 64 + 4 pad: half-wave K-phases hit disjoint bank groups

// ---------------------------------------------------------------------------
// Kernel 0: zero the accumulation workspace (gram partials + row sums)
// ---------------------------------------------------------------------------
__global__ void fioU_zero_ws(float* __restrict__ ws, int n) {
    int i = blockIdx.x * blockDim.x + threadIdx.x;
    if (i < n) ws[i] = 0.0f;
}

// ---------------------------------------------------------------------------
// Kernel 1: batched Gram via V_WMMA_F32_16X16X4_F32 + row sums.
// grid = NB * KC workgroups, 512 threads (16 waves) each.
// Wave w computes the 16x16 tile (ti = 16*(w>>2), tj = 16*(w&3)).
// ---------------------------------------------------------------------------
__global__ __launch_bounds__(512)
void fioU_gram_kernel(const float* __restrict__ masks,
                      float* __restrict__ gram,   // [NB][NN][NN]
                      float* __restrict__ sums)   // [NB][NN]
{
    __shared__ float smem[NN * LDS_STRIDE];   // 64 rows x 68 floats = 17408 B

    const int b    = blockIdx.x & (NB - 1);
    const int kc   = blockIdx.x >> 5;
    const int t    = threadIdx.x;
    const int wave = t >> 5;
    const int lane = t & 31;
    const int r    = lane & 15;     // row/col within tile
    const int half = lane >> 4;     // K phase (0: k,k+1  1: k+2,k+3)
    const int ti   = (wave >> 2) * 16;
    const int tj   = (wave & 3) * 16;

    // global->LDS loader mapping: 8 threads per row, 8 floats per thread
    const int lrow = t >> 3;
    const int lcol = (t & 7) * 8;

    const float* src = masks + ((size_t)b * NN + lrow) * HW
                             + (size_t)kc * KCHUNK + lcol;

    v8f  acc  = {0.f, 0.f, 0.f, 0.f, 0.f, 0.f, 0.f, 0.f};
    float ssum = 0.0f;

    for (int ks = 0; ks < KCHUNK; ks += KS) {
        // ---- stage: load, clip ((x+1)/2 clamped at 0), accumulate row sum
        const float* p = src + ks;
        float x[8];
        #pragma unroll
        for (int j = 0; j < 8; ++j) {
            float v = p[j];
            v = fmaxf(fmaf(v, 0.5f, 0.5f), 0.0f);
            x[j] = v;
            ssum += v;
        }
        __syncthreads();   // previous iteration's LDS readers are done
        #pragma unroll
        for (int j = 0; j < 8; ++j)
            smem[lrow * LDS_STRIDE + lcol + j] = x[j];
        __syncthreads();   // staged data visible to all waves

        // ---- compute: 16 WMMAs sweep K = ks .. ks+63 in steps of 4
        #pragma unroll
        for (int kk = 0; kk < KS; kk += 4) {
            const int col = kk + 2 * half;
            v2f a  = *(const v2f*)&smem[(ti + r) * LDS_STRIDE + col];
            v2f bm = *(const v2f*)&smem[(tj + r) * LDS_STRIDE + col];
            // D = A(16x4, rows ti..) x B(4x16, cols tj..) + C   (fp32)
            acc = __builtin_amdgcn_wmma_f32_16x16x4_f32(
                      /*neg_a=*/false, a, /*neg_b=*/false, bm,
                      /*c_mod=*/(short)0, acc,
                      /*reuse_a=*/false, /*reuse_b=*/false);
        }
    }

    // ---- commit gram partial: C/D layout => element (v + 8*half, r)
    float* g = gram + ((size_t)b * NN + ti) * NN + tj;
    #pragma unroll
    for (int v = 0; v < 8; ++v)
        atomicAdd(&g[(size_t)(v + 8 * half) * NN + r], acc[v]);

    // ---- commit row sums: reduce 8 loader-partials per row
    __syncthreads();
    smem[t] = ssum;
    __syncthreads();
    if (t < NN) {
        float stot = 0.0f;
        #pragma unroll
        for (int j = 0; j < 8; ++j) stot += smem[t * 8 + j];
        atomicAdd(&sums[b * NN + t], stot);
    }
}

// ---------------------------------------------------------------------------
// Kernel 2: cr = mean_b(G / min(s_i,s_j)); node-type masked mean-abs loss.
// ---------------------------------------------------------------------------
__global__ __launch_bounds__(256)
void fioU_finalize(const float* __restrict__ gram,
                   const float* __restrict__ sums,
                   const int*   __restrict__ nodes,
                   float* __restrict__ out)
{
    __shared__ float red_l[256];
    __shared__ float red_w[256];
    __shared__ float s_s[NB * NN];
    __shared__ int   s_t[NN];

    const int t = threadIdx.x;
    for (int i = t; i < NB * NN; i += 256) s_s[i] = sums[i];
    if (t < NN) {
        int v = nodes[t];
        s_t[t] = (v < 7) ? 0 : ((v < 9) ? 1 : 2);
    }
    __syncthreads();

    float lsum = 0.0f, wsum = 0.0f;
    for (int p = t; p < NN * NN; p += 256) {
        const int i = p >> 6, j = p & 63;
        if (i < j) {
            const int a = s_t[i], c = s_t[j];
            const bool has_f = (a == 1) | (c == 1);
            const bool has_a = (a == 2) | (c == 2);
            if (!(has_f && !has_a)) {
                const float beta = ((a == 2) != (c == 2)) ? 1.0f : 0.0f;
                float accv = 0.0f;
                for (int b = 0; b < NB; ++b) {
                    const float g  = gram[((size_t)b * NN + i) * NN + j];
                    const float mn = fminf(s_s[b * NN + i], s_s[b * NN + j]);
                    accv += g / mn;
                }
                lsum += fabsf(beta - accv * (1.0f / NB));
                wsum += 1.0f;
            }
        }
    }
    red_l[t] = lsum; red_w[t] = wsum;
    __syncthreads();
    for (int s = 128; s > 0; s >>= 1) {
        if (t < s) { red_l[t] += red_l[t + s]; red_w[t] += red_w[t + s]; }
        __syncthreads();
    }
    if (t == 0) out[0] = red_l[0] / red_w[0];
}

// ---------------------------------------------------------------------------
extern "C" void kernel_launch(void* const* d_in, const int* in_sizes, int n_in,
                              void* d_out, int out_size, void* d_ws, size_t ws_size,
                              hipStream_t stream)
{
    const float* masks = (const float*)d_in[0];
    const int*   nodes = (const int*)d_in[1];

    float* gram = (float*)d_ws;                       // NB*NN*NN floats
    float* sums = gram + (size_t)NB * NN * NN;        // NB*NN floats
    float* out  = (float*)d_out;

    const int nzero = NB * NN * NN + NB * NN;
    fioU_zero_ws<<<(nzero + 255) / 256, 256, 0, stream>>>(gram, nzero);
    fioU_gram_kernel<<<NB * KC, 512, 0, stream>>>(masks, gram, sums);
    fioU_finalize<<<1, 256, 0, stream>>>(gram, sums, nodes, out);
}